// MinGRU_3779571220572
// MI455X (gfx1250) — compile-verified
//
#include <hip/hip_runtime.h>

typedef __attribute__((ext_vector_type(2))) float v2f;
typedef __attribute__((ext_vector_type(8))) float v8f;

#define B_    4
#define S_    8192
#define D_    1024
#define TWO_D 2048
#define CHUNK 256
#define NCH   (S_ / CHUNK)   // 32

__device__ __forceinline__ float softplus_f(float x) {
    // stable: max(x,0) + log1p(exp(-|x|))
    return fmaxf(x, 0.0f) + log1pf(__expf(-fabsf(x)));
}
__device__ __forceinline__ float log_g_f(float x) {
    // x>=0: log(relu(x)+0.5) = log(x+0.5); else -softplus(-x)
    return (x >= 0.0f) ? __logf(x + 0.5f) : -softplus_f(-x);
}
__device__ __forceinline__ float logaddexp_f(float a, float b) {
    float m = fmaxf(a, b);
    float n = fminf(a, b);
    if (m <= -1e30f) return m;            // both -inf
    return m + log1pf(__expf(n - m));
}

// ---------------------------------------------------------------------------
// Phase 1: fused GEMM (x @ W) + log-space epilogue via V_WMMA_F32_16X16X4_F32.
// Each wave owns one 16x16 hidden tile AND the matching gate tile (col+1024),
// sharing the A fragment. Writes lc -> ws, lv -> d_out (consumed in-place).
// ---------------------------------------------------------------------------
__global__ __launch_bounds__(256) void mingru_gemm_wmma(
    const float* __restrict__ x, const float* __restrict__ W,
    const long long* __restrict__ ids,
    float* __restrict__ lc_out, float* __restrict__ lv_out)
{
    const int lane = threadIdx.x & 31;
    const int wave = threadIdx.x >> 5;
    const int tile = blockIdx.x * 8 + wave;       // 2048 row tiles * 64 col tiles
    const int m0 = (tile >> 6) << 4;              // row base
    const int n0 = (tile & 63) << 4;              // hidden col base
    const int lg = lane >> 4;                     // lane half-group: 0 or 1
    const int lr = lane & 15;

    // A fragment: lane holds x[m0+lr, k0+2*lg .. +1]  (8B contiguous load)
    const float* xrow = x + (size_t)(m0 + lr) * D_ + 2 * lg;
    // B fragments: lane holds W[k0+2*lg .. +1, n0+lr]
    const float* wh = W + (size_t)(2 * lg) * TWO_D + n0 + lr;
    const float* wg = wh + D_;

    v8f acc_h = {};
    v8f acc_g = {};
    #pragma unroll 4
    for (int k0 = 0; k0 < D_; k0 += 4) {
        v2f a = *(const v2f*)(xrow + k0);
        const float* p = wh + (size_t)k0 * TWO_D;
        v2f bh; bh.x = p[0]; bh.y = p[TWO_D];
        p = wg + (size_t)k0 * TWO_D;
        v2f bg; bg.x = p[0]; bg.y = p[TWO_D];
        acc_h = __builtin_amdgcn_wmma_f32_16x16x4_f32(
            false, a, false, bh, (short)0, acc_h, false, false);
        acc_g = __builtin_amdgcn_wmma_f32_16x16x4_f32(
            false, a, false, bg, (short)0, acc_g, false, false);
    }

    // Epilogue: C/D layout -> VGPR i: M = i + 8*lg, N = lane&15
    #pragma unroll
    for (int i = 0; i < 8; ++i) {
        int row = m0 + i + 8 * lg;                // global (b*S + s)
        int col = n0 + lr;                        // d
        int b = row >> 13;                        // / S_
        int s = row & (S_ - 1);
        float h = acc_h[i];
        float g = acc_g[i];
        float lc = -softplus_f(g);                       // log(1-z)
        float lv = -softplus_f(-g) + log_g_f(h);         // log z + log g(h)
        bool reset = (s > 0) && (ids[(size_t)b * S_ + s - 1] == 0);
        if (reset) { lc = -100.0f; lv = -100.0f; }
        size_t o = (size_t)row * D_ + col;
        lc_out[o] = lc;
        lv_out[o] = lv;
    }
}

// ---------------------------------------------------------------------------
// Phase 2: per-chunk composition (A, V) of the log-space linear recurrence.
// Thread layout keeps d fastest -> fully coalesced across lanes.
// ---------------------------------------------------------------------------
__global__ __launch_bounds__(256) void mingru_chunk_reduce(
    const float* __restrict__ lc, const float* __restrict__ lv,
    float* __restrict__ Ac, float* __restrict__ Vc)
{
    int tid = blockIdx.x * 256 + threadIdx.x;     // ((b*NCH + c)*D + d)
    int d  = tid & (D_ - 1);
    int bc = tid >> 10;
    int c  = bc & (NCH - 1);
    int b  = bc >> 5;
    size_t base = ((size_t)b * S_ + (size_t)c * CHUNK) * D_ + d;
    float A = 0.0f, V = -INFINITY;
    for (int t = 0; t < CHUNK; ++t) {
        float c1 = lc[base + (size_t)t * D_];
        float v1 = lv[base + (size_t)t * D_];
        V = logaddexp_f(V + c1, v1);
        A += c1;
    }
    Ac[tid] = A;
    Vc[tid] = V;
}

// ---------------------------------------------------------------------------
// Phase 3: tiny inter-chunk sequential scan (B*D threads, NCH steps).
// Produces the incoming log_h for every chunk.
// ---------------------------------------------------------------------------
__global__ __launch_bounds__(256) void mingru_chunk_scan(
    const float* __restrict__ Ac, const float* __restrict__ Vc,
    float* __restrict__ Hin)
{
    int tid = blockIdx.x * 256 + threadIdx.x;     // b*D + d
    int d = tid & (D_ - 1);
    int b = tid >> 10;
    float H = -INFINITY;
    for (int c = 0; c < NCH; ++c) {
        int idx = ((b * NCH + c) << 10) + d;
        Hin[idx] = H;
        H = logaddexp_f(Ac[idx] + H, Vc[idx]);
    }
}

// ---------------------------------------------------------------------------
// Phase 4: apply within-chunk scan with carry-in; out = exp(log_h), written
// in place over lv (stored in d_out). Last chunk also emits next_prev_hidden.
// ---------------------------------------------------------------------------
__global__ __launch_bounds__(256) void mingru_apply(
    const float* __restrict__ lc, float* __restrict__ out,
    const float* __restrict__ Hin, const long long* __restrict__ ids,
    float* __restrict__ next_h)
{
    int tid = blockIdx.x * 256 + threadIdx.x;
    int d  = tid & (D_ - 1);
    int bc = tid >> 10;
    int c  = bc & (NCH - 1);
    int b  = bc >> 5;
    size_t base = ((size_t)b * S_ + (size_t)c * CHUNK) * D_ + d;
    float H = Hin[tid];
    for (int t = 0; t < CHUNK; ++t) {
        size_t o = base + (size_t)t * D_;
        float c1 = lc[o];
        float v1 = out[o];                        // lv written by phase 1
        H = logaddexp_f(c1 + H, v1);
        out[o] = __expf(H);
    }
    if (c == NCH - 1) {
        bool pad = (ids[(size_t)b * S_ + S_ - 1] == 0);
        next_h[(size_t)b * D_ + d] = pad ? 0.0f : __expf(H);
    }
}

extern "C" void kernel_launch(void* const* d_in, const int* in_sizes, int n_in,
                              void* d_out, int out_size, void* d_ws, size_t ws_size,
                              hipStream_t stream) {
    (void)in_sizes; (void)n_in; (void)out_size; (void)ws_size;
    const float*     x   = (const float*)d_in[0];
    const float*     W   = (const float*)d_in[1];
    const long long* ids = (const long long*)d_in[2];

    float* out    = (float*)d_out;                       // [B,S,D] then [B,1,D]
    float* lv     = out;                                 // lv staged in out region
    float* next_h = out + (size_t)B_ * S_ * D_;

    float* lc  = (float*)d_ws;                           // B*S*D floats (128 MB)
    float* Ac  = lc + (size_t)B_ * S_ * D_;              // B*NCH*D
    float* Vc  = Ac + (size_t)B_ * NCH * D_;
    float* Hin = Vc + (size_t)B_ * NCH * D_;

    const int gemm_blocks = (2048 * 64) / 8;             // 8 wave-tiles per block
    mingru_gemm_wmma<<<gemm_blocks, 256, 0, stream>>>(x, W, ids, lc, lv);

    const int nthreads = B_ * NCH * D_;                  // 131072
    mingru_chunk_reduce<<<nthreads / 256, 256, 0, stream>>>(lc, lv, Ac, Vc);
    mingru_chunk_scan<<<(B_ * D_) / 256, 256, 0, stream>>>(Ac, Vc, Hin);
    mingru_apply<<<nthreads / 256, 256, 0, stream>>>(lc, out, Hin, ids, next_h);
}